// ValueNetwork_68453188764140
// MI455X (gfx1250) — compile-verified
//
#include <hip/hip_runtime.h>

// MI455X / gfx1250 — wave32, WMMA (V_WMMA_F32_16X16X4_F32) fused value-network.
// One wave per batch element, 4 waves (128 threads) per block.
//
// Exploits: complete-graph segment_sum == colsum - self; only node 0 of the
// second gconv is consumed; all dense matmuls run as exact-fp32 WMMA tiles.
// All pad-masking is branch-free (clamped address + arithmetic mask) so EXEC
// stays all-ones and no saveexec clusters appear in the WMMA loops.

typedef float v2f __attribute__((ext_vector_type(2)));
typedef float v8f __attribute__((ext_vector_type(8)));

#define B_TOT    4096
#define NH_      31
#define IN_DIM_  13
#define SELF_D   6
#define HUM_D    7
#define NN_      32      // graph nodes
#define XD_      32      // node feature dim
#define MLP_H    64      // MLP hidden
#define GC_H     52      // gconv hidden
#define WPB      4       // waves per block

__device__ __forceinline__ float relu_(float x) { return fmaxf(x, 0.0f); }

__device__ __forceinline__ v8f wmma4(v8f c, v2f a, v2f b) {
  // D = A(16x4,f32) * B(4x16,f32) + C(16x16,f32)
  return __builtin_amdgcn_wmma_f32_16x16x4_f32(
      /*neg_a=*/false, a, /*neg_b=*/false, b,
      /*c_mod=*/(short)0, c, /*reuse_a=*/false, /*reuse_b=*/false);
}

__global__ __launch_bounds__(WPB * 32)
void vnet_fused_kernel(
    const float* __restrict__ state,
    const float* __restrict__ wr_w1, const float* __restrict__ wr_b1,
    const float* __restrict__ wr_w2, const float* __restrict__ wr_b2,
    const float* __restrict__ wh_w1, const float* __restrict__ wh_b1,
    const float* __restrict__ wh_w2, const float* __restrict__ wh_b2,
    const float* __restrict__ c1_rel_w, const float* __restrict__ c1_rel_b,
    const float* __restrict__ c1_root_w,
    const float* __restrict__ c2_rel_w, const float* __restrict__ c2_rel_b,
    const float* __restrict__ c2_root_w,
    const float* __restrict__ vn_w1, const float* __restrict__ vn_b1,
    const float* __restrict__ vn_w2, const float* __restrict__ vn_b2,
    const float* __restrict__ vn_w3, const float* __restrict__ vn_b3,
    float* __restrict__ out)
{
  __shared__ float sHF [WPB][NN_][9];       // zero-padded human features (32x8)
  __shared__ float sX  [WPB][NN_][33];      // node features X (32x32)
  __shared__ float sBuf[WPB][NN_ * 65];     // overlay: MLP hidden (stride 65) -> gconv1 out H1 (stride 53)
  __shared__ float sS  [WPB][64];           // overlay: self hidden(64) -> X colsum(32)
  __shared__ float sS2 [WPB][56];           // H1 colsum (52)
  __shared__ float sH0 [WPB][32];           // node-0 features after gconv2
  __shared__ float sV1 [WPB][152];
  __shared__ float sV2 [WPB][104];

  const int w = threadIdx.x >> 5;           // wave in block
  const int l = threadIdx.x & 31;           // lane
  const int b = blockIdx.x * WPB + w;       // batch element
  const float* st = state + (size_t)b * NH_ * IN_DIM_;

  // WMMA f32 16x16x4 lane decomposition (wave32):
  const int lo   = l & 15;                  // row (A) / col (B,C,D) within tile
  const int koff = (l >> 4) << 1;           // K sub-offset: 0 or 2
  const int hi8  = (l >> 4) << 3;           // C/D row offset: 0 or 8

  // ---------- stage human features into zero-padded LDS (branch-free mask) ----------
  #pragma unroll
  for (int t = 0; t < 8; ++t) {             // 256 entries / 32 lanes
    const int idx = l + 32 * t;
    const int row = idx >> 3;
    const int k   = idx & 7;
    const int rc  = row < NH_ ? row : NH_ - 1;   // clamped, always legal
    const int kc  = k < HUM_D ? k : HUM_D - 1;
    const float m = (row < NH_ && k < HUM_D) ? 1.0f : 0.0f;
    sHF[w][row][k] = m * st[rc * IN_DIM_ + SELF_D + kc];
  }

  // ---------- self path, layer 1 (VALU): 6 -> 64 ----------
  {
    float sf[SELF_D];
    #pragma unroll
    for (int i = 0; i < SELF_D; ++i) sf[i] = st[i];
    #pragma unroll
    for (int t = 0; t < 2; ++t) {
      int j = l + 32 * t;
      float acc = wr_b1[j];
      #pragma unroll
      for (int i = 0; i < SELF_D; ++i) acc += sf[i] * wr_w1[i * 64 + j];
      sS[w][j] = relu_(acc);
    }
  }
  __syncthreads();

  // ---------- self path, layer 2 (VALU): 64 -> 32 => X[0] ----------
  {
    float acc = wr_b2[l];
    for (int i = 0; i < MLP_H; ++i) acc += sS[w][i] * wr_w2[i * 32 + l];
    sX[w][0][l] = relu_(acc);
  }

  // ---------- human MLP layer 1 (WMMA): A=human(32x8 padded), B=wh_w1(7x64 pad 8x64) ----------
  for (int mt = 0; mt < 32; mt += 16) {
    for (int nt = 0; nt < MLP_H; nt += 16) {
      v8f c = {0.f, 0.f, 0.f, 0.f, 0.f, 0.f, 0.f, 0.f};
      #pragma unroll
      for (int ks = 0; ks < 8; ks += 4) {
        const int row = mt + lo;
        const int kk  = ks + koff;
        const int col = nt + lo;
        v2f a, bb;
        a.x = sHF[w][row][kk];
        a.y = sHF[w][row][kk + 1];
        // branch-free K pad: clamp address, mask value
        const float m0 = (kk     < HUM_D) ? 1.0f : 0.0f;
        const float m1 = (kk + 1 < HUM_D) ? 1.0f : 0.0f;
        const int k0 = kk     < HUM_D ? kk     : HUM_D - 1;
        const int k1 = kk + 1 < HUM_D ? kk + 1 : HUM_D - 1;
        bb.x = m0 * wh_w1[k0 * MLP_H + col];
        bb.y = m1 * wh_w1[k1 * MLP_H + col];
        c = wmma4(c, a, bb);
      }
      const int col = nt + lo;
      const float bias = wh_b1[col];
      #pragma unroll
      for (int r = 0; r < 8; ++r)
        sBuf[w][(mt + hi8 + r) * 65 + col] = relu_(c[r] + bias);
    }
  }
  __syncthreads();

  // ---------- human MLP layer 2 (WMMA): A=hidden(32x64), B=wh_w2(64x32) => X[1..31] ----------
  for (int mt = 0; mt < 32; mt += 16) {
    for (int nt = 0; nt < XD_; nt += 16) {
      v8f c = {0.f, 0.f, 0.f, 0.f, 0.f, 0.f, 0.f, 0.f};
      for (int ks = 0; ks < MLP_H; ks += 4) {
        const int kk = ks + koff;
        const int col = nt + lo;
        v2f a, bb;
        a.x  = sBuf[w][(mt + lo) * 65 + kk];
        a.y  = sBuf[w][(mt + lo) * 65 + kk + 1];
        bb.x = wh_w2[kk * XD_ + col];
        bb.y = wh_w2[(kk + 1) * XD_ + col];
        c = wmma4(c, a, bb);
      }
      const int col = nt + lo;
      const float bias = wh_b2[col];
      #pragma unroll
      for (int r = 0; r < 8; ++r) {
        const int row = mt + hi8 + r;          // human index
        if (row < NH_) sX[w][row + 1][col] = relu_(c[r] + bias);
      }
    }
  }
  __syncthreads();

  // ---------- column sum of X (segment_sum over complete graph) ----------
  {
    float s = 0.f;
    #pragma unroll
    for (int n = 0; n < NN_; ++n) s += sX[w][n][l];
    sS[w][l] = s;
  }
  __syncthreads();

  // ---------- gconv1 (WMMA): relu((colsum-X)@rel_w + X@root_w + b), 32 -> 52 ----------
  for (int mt = 0; mt < 32; mt += 16) {
    const int row = mt + lo;
    for (int nt = 0; nt < 64; nt += 16) {
      const int col  = nt + lo;
      const int colc = col < GC_H ? col : GC_H - 1;   // clamped, always legal
      const float cm = (col < GC_H) ? 1.0f : 0.0f;    // branch-free N pad mask
      v8f c = {0.f, 0.f, 0.f, 0.f, 0.f, 0.f, 0.f, 0.f};
      for (int ks = 0; ks < XD_; ks += 4) {    // agg @ c1_rel_w
        const int kk = ks + koff;
        v2f a, bb;
        a.x  = sS[w][kk]     - sX[w][row][kk];
        a.y  = sS[w][kk + 1] - sX[w][row][kk + 1];
        bb.x = cm * c1_rel_w[kk * GC_H + colc];
        bb.y = cm * c1_rel_w[(kk + 1) * GC_H + colc];
        c = wmma4(c, a, bb);
      }
      for (int ks = 0; ks < XD_; ks += 4) {    // X @ c1_root_w (accumulate)
        const int kk = ks + koff;
        v2f a, bb;
        a.x  = sX[w][row][kk];
        a.y  = sX[w][row][kk + 1];
        bb.x = cm * c1_root_w[kk * GC_H + colc];
        bb.y = cm * c1_root_w[(kk + 1) * GC_H + colc];
        c = wmma4(c, a, bb);
      }
      if (col < GC_H) {
        const float bias = c1_rel_b[col];
        #pragma unroll
        for (int r = 0; r < 8; ++r)
          sBuf[w][(mt + hi8 + r) * 53 + col] = relu_(c[r] + bias);   // H1, stride 53
      }
    }
  }
  __syncthreads();

  // ---------- column sum of H1 ----------
  for (int j = l; j < GC_H; j += 32) {
    float s = 0.f;
    #pragma unroll
    for (int n = 0; n < NN_; ++n) s += sBuf[w][n * 53 + j];
    sS2[w][j] = s;
  }
  __syncthreads();

  // ---------- gconv2, node 0 only (VALU): 52 -> 32 ----------
  {
    float acc = c2_rel_b[l];
    for (int k = 0; k < GC_H; ++k) {
      const float hk = sBuf[w][k];             // H1[0][k]
      acc += (sS2[w][k] - hk) * c2_rel_w[k * 32 + l] + hk * c2_root_w[k * 32 + l];
    }
    sH0[w][l] = relu_(acc);
  }
  __syncthreads();

  // ---------- value head (VALU): 32 -> 150 -> 100 -> 1 ----------
  for (int j = l; j < 150; j += 32) {
    float acc = vn_b1[j];
    #pragma unroll
    for (int i = 0; i < 32; ++i) acc += sH0[w][i] * vn_w1[i * 150 + j];
    sV1[w][j] = relu_(acc);
  }
  __syncthreads();
  for (int j = l; j < 100; j += 32) {
    float acc = vn_b2[j];
    for (int i = 0; i < 150; ++i) acc += sV1[w][i] * vn_w2[i * 100 + j];
    sV2[w][j] = relu_(acc);
  }
  __syncthreads();
  {
    float p = 0.f;
    for (int i = l; i < 100; i += 32) p += sV2[w][i] * vn_w3[i];
    #pragma unroll
    for (int off = 16; off > 0; off >>= 1) p += __shfl_down(p, off, 32);
    if (l == 0) out[b] = p + vn_b3[0];
  }
}

extern "C" void kernel_launch(void* const* d_in, const int* in_sizes, int n_in,
                              void* d_out, int out_size, void* d_ws, size_t ws_size,
                              hipStream_t stream) {
  (void)in_sizes; (void)n_in; (void)d_ws; (void)ws_size; (void)out_size;
  const float* state    = (const float*)d_in[0];
  // d_in[1]=src, d_in[2]=dst: complete-graph edges, algebraically folded away.
  const float* wr_w1    = (const float*)d_in[3];
  const float* wr_b1    = (const float*)d_in[4];
  const float* wr_w2    = (const float*)d_in[5];
  const float* wr_b2    = (const float*)d_in[6];
  const float* wh_w1    = (const float*)d_in[7];
  const float* wh_b1    = (const float*)d_in[8];
  const float* wh_w2    = (const float*)d_in[9];
  const float* wh_b2    = (const float*)d_in[10];
  const float* c1_rel_w = (const float*)d_in[11];
  const float* c1_rel_b = (const float*)d_in[12];
  const float* c1_root_w= (const float*)d_in[13];
  const float* c2_rel_w = (const float*)d_in[14];
  const float* c2_rel_b = (const float*)d_in[15];
  const float* c2_root_w= (const float*)d_in[16];
  const float* vn_w1    = (const float*)d_in[17];
  const float* vn_b1    = (const float*)d_in[18];
  const float* vn_w2    = (const float*)d_in[19];
  const float* vn_b2    = (const float*)d_in[20];
  const float* vn_w3    = (const float*)d_in[21];
  const float* vn_b3    = (const float*)d_in[22];
  // d_in[23] = dropout (0) : ignored.

  dim3 grid(B_TOT / WPB);
  dim3 block(WPB * 32);
  vnet_fused_kernel<<<grid, block, 0, stream>>>(
      state, wr_w1, wr_b1, wr_w2, wr_b2, wh_w1, wh_b1, wh_w2, wh_b2,
      c1_rel_w, c1_rel_b, c1_root_w, c2_rel_w, c2_rel_b, c2_root_w,
      vn_w1, vn_b1, vn_w2, vn_b2, vn_w3, vn_b3, (float*)d_out);
}